// ParticleProjection_72576357367816
// MI455X (gfx1250) — compile-verified
//
#include <hip/hip_runtime.h>
#include <math.h>

#define CAM_FL 540.0f
#define CAM_W  640
#define CAM_H  480

typedef __attribute__((ext_vector_type(2))) float v2f;
typedef __attribute__((ext_vector_type(8))) float v8f;

// One thread per particle; 256 threads/block = 8 wave32; grid.y = batch.
// Transform of 16 particles per V_WMMA_F32_16X16X4_F32: D = A(16x4) * B(4x16)
//   A row m = [x_m, y_m, z_m, 1],  B cols 0..2 = [R | t] (cols 3..15 zero)
__global__ __launch_bounds__(256) void particle_splat_kernel(
    const float* __restrict__ locs,         // (B,N,3)
    const float* __restrict__ camera_pose,  // (B,3)
    const float* __restrict__ camera_rot,   // (B,4)
    const float* __restrict__ depth,        // (B,480,640)
    float*       __restrict__ out,          // (B,480,640)
    int N)
{
    const int b        = blockIdx.y;
    const int tid      = threadIdx.x;          // 0..255
    const int lane     = tid & 31;             // wave32 lane
    const int waveBase = tid & ~31;            // first slot of this wave
    const int p        = blockIdx.x * 256 + tid;
    const bool active  = (p < N);

    // ---------- per-batch transform (uniform across block) ----------
    float qx = camera_rot[b * 4 + 0];
    float qy = camera_rot[b * 4 + 1];
    float qz = camera_rot[b * 4 + 2];
    float qw = camera_rot[b * 4 + 3];
    float inv = 1.0f / sqrtf(qx * qx + qy * qy + qz * qz + qw * qw);
    qx = -qx * inv; qy = -qy * inv; qz = -qz * inv; qw = qw * inv; // conjugate

    const float r00 = 1.f - 2.f * (qy * qy + qz * qz);
    const float r01 = 2.f * (qx * qy + qz * qw);
    const float r02 = 2.f * (qx * qz - qy * qw);
    const float r10 = 2.f * (qx * qy - qz * qw);
    const float r11 = 1.f - 2.f * (qx * qx + qz * qz);
    const float r12 = 2.f * (qy * qz + qx * qw);
    const float r20 = 2.f * (qx * qz + qy * qw);
    const float r21 = 2.f * (qy * qz - qx * qw);
    const float r22 = 1.f - 2.f * (qx * qx + qy * qy);

    const float c0 = camera_pose[b * 3 + 0];
    const float c1 = camera_pose[b * 3 + 1];
    const float c2 = camera_pose[b * 3 + 2];
    const float t0 = -(c0 * r00 + c1 * r10 + c2 * r20);
    const float t1 = -(c0 * r01 + c1 * r11 + c2 * r21);
    const float t2 = -(c0 * r02 + c1 * r12 + c2 * r22);

    // ---------- build B operand (4x16 f32): v0 = rows K0|K2, v1 = rows K1|K3 ----------
    const int  col = lane & 15;        // N column of the WMMA tile
    const bool hi  = (lane >= 16);
    const float row0 = (col == 0) ? r00 : (col == 1) ? r01 : (col == 2) ? r02 : 0.f;
    const float row1 = (col == 0) ? r10 : (col == 1) ? r11 : (col == 2) ? r12 : 0.f;
    const float row2 = (col == 0) ? r20 : (col == 1) ? r21 : (col == 2) ? r22 : 0.f;
    const float row3 = (col == 0) ? t0  : (col == 1) ? t1  : (col == 2) ? t2  : 0.f;
    v2f bm;
    bm.x = hi ? row2 : row0;
    bm.y = hi ? row3 : row1;

    // ---------- stage particle coords in LDS ----------
    __shared__ float sX[256], sY[256], sZ[256];
    __shared__ float sOX[256], sOY[256], sOZ[256];

    float lx = 0.f, ly = 0.f, lz = 0.f;   // tail particles: harmless finite dummies
    if (active) {
        const float* lp = locs + ((long long)b * N + p) * 3;
        lx = lp[0]; ly = lp[1]; lz = lp[2];
    }
    sX[tid] = lx; sY[tid] = ly; sZ[tid] = lz;
    __syncthreads();

    // ---------- two 16-particle WMMA tiles per wave ----------
#pragma unroll
    for (int tau = 0; tau < 2; ++tau) {
        const int pp = waveBase + tau * 16 + col;   // block-local particle slot
        v2f a;
        // A layout 16x4 f32: lanes0-15 -> K0/K1 (x,y); lanes16-31 -> K2/K3 (z,1)
        a.x = hi ? sZ[pp] : sX[pp];
        a.y = hi ? 1.0f   : sY[pp];
        v8f c = {0.f, 0.f, 0.f, 0.f, 0.f, 0.f, 0.f, 0.f};
        v8f d = __builtin_amdgcn_wmma_f32_16x16x4_f32(
            /*neg_a=*/false, a, /*neg_b=*/false, bm,
            /*c_mod=*/(short)0, c, /*reuse_a=*/false, /*reuse_b=*/false);

        // D layout 16x16 f32: lane = column N, VGPR m = row M (lanes>=16 -> M+8)
        const int sbase = waveBase + tau * 16 + (hi ? 8 : 0);
        if (col == 0) {
#pragma unroll
            for (int m = 0; m < 8; ++m) sOX[sbase + m] = d[m];
        } else if (col == 1) {
#pragma unroll
            for (int m = 0; m < 8; ++m) sOY[sbase + m] = d[m];
        } else if (col == 2) {
#pragma unroll
            for (int m = 0; m < 8; ++m) sOZ[sbase + m] = d[m];
        }
    }
    __syncthreads();

    const float X = sOX[tid];
    const float Y = sOY[tid];
    const float Z = sOZ[tid];

    if (!active) return;
    if (!(Z > 0.f)) return;               // depth test always fails otherwise

    // ---------- projection ----------
    const float zinv = 1.0f / Z;
    const float fpx = X * zinv * CAM_FL + (float)CAM_W * 0.5f;
    const float fpy = Y * zinv * CAM_FL + (float)CAM_H * 0.5f;
    const float fx = floorf(fpx);
    const float fy = floorf(fpy);
    // window [fx-2, fx+2] must intersect [0, W-1]; float-domain check avoids int overflow
    if (!(fx >= -2.f && fx <= (float)(CAM_W + 1) &&
          fy >= -2.f && fy <= (float)(CAM_H + 1))) return;
    const int jx0 = (int)fx;
    const int iy0 = (int)fy;

    // ---------- separable 5-tap Gaussian weights (std=1, scale=1) ----------
    float wx[5], wy[5];
#pragma unroll
    for (int o = 0; o < 5; ++o) {
        const float dx = fx + (float)(o - 2) - fpx;
        const float dy = fy + (float)(o - 2) - fpy;
        wx[o] = __expf(-0.5f * dx * dx);
        wy[o] = __expf(-0.5f * dy * dy);
    }

    const float* drow0 = depth + (long long)b * CAM_H * CAM_W;
    float*       orow0 = out   + (long long)b * CAM_H * CAM_W;
#pragma unroll
    for (int oy = 0; oy < 5; ++oy) {
        const int ii = iy0 + oy - 2;
        if ((unsigned)ii >= (unsigned)CAM_H) continue;
        const float* drow = drow0 + (long long)ii * CAM_W;
        float*       orow = orow0 + (long long)ii * CAM_W;
#pragma unroll
        for (int ox = 0; ox < 5; ++ox) {
            const int jj = jx0 + ox - 2;
            if ((unsigned)jj >= (unsigned)CAM_W) continue;
            const float dval = drow[jj];
            if (Z <= dval) {
                atomicAdd(&orow[jj], wy[oy] * wx[ox]);  // L2-resident f32 atomic
            }
        }
    }
}

extern "C" void kernel_launch(void* const* d_in, const int* in_sizes, int n_in,
                              void* d_out, int out_size, void* d_ws, size_t ws_size,
                              hipStream_t stream) {
    const float* locs  = (const float*)d_in[0];   // (B,N,3) f32
    const float* pose  = (const float*)d_in[1];   // (B,3)   f32
    const float* rot   = (const float*)d_in[2];   // (B,4)   f32
    const float* depth = (const float*)d_in[3];   // (B,480,640) f32
    float* out = (float*)d_out;                   // (B,480,640) f32

    const int B = in_sizes[1] / 3;                // camera_pose is (B,3)
    const int N = in_sizes[0] / (3 * B);          // locs is (B,N,3)

    // harness poisons d_out; zero it every launch (graph-capture-safe)
    hipMemsetAsync(out, 0, (size_t)out_size * sizeof(float), stream);

    dim3 grid((unsigned)((N + 255) / 256), (unsigned)B, 1);
    particle_splat_kernel<<<grid, dim3(256, 1, 1), 0, stream>>>(
        locs, pose, rot, depth, out, N);
}